// MVPPrompt_6914897346762
// MI455X (gfx1250) — compile-verified
//
#include <hip/hip_runtime.h>
#include <hip/hip_bf16.h>

#define B_SZ  2048
#define POOL  512
#define TOPK  4
#define EPL   8
#define EMB   768
#define KEYD  768

typedef __attribute__((ext_vector_type(2))) float v2f;
typedef __attribute__((ext_vector_type(8))) float v8f;

__device__ __forceinline__ float wave_sum(float v) {
#pragma unroll
  for (int off = 16; off; off >>= 1) v += __shfl_xor(v, off, 32);
  return v;
}

// L2-normalize 768-long rows; one wave32 per row, 8 rows per 256-thread block.
__global__ void l2norm_rows(const float* __restrict__ in, float* __restrict__ out) {
  int wave = threadIdx.x >> 5;
  int lane = threadIdx.x & 31;
  int row  = blockIdx.x * (blockDim.x >> 5) + wave;
  const float* src = in  + (size_t)row * KEYD;
  float*       dst = out + (size_t)row * KEYD;
  float vals[KEYD / 32];
  float s = 0.f;
#pragma unroll
  for (int t = 0; t < KEYD / 32; ++t) {
    float v = src[lane + 32 * t];
    vals[t] = v;
    s += v * v;
  }
  s = wave_sum(s);
  float inv = 1.f / fmaxf(sqrtf(s), 1e-12f);
#pragma unroll
  for (int t = 0; t < KEYD / 32; ++t) dst[lane + 32 * t] = vals[t] * inv;
}

// cos[b][p] = q_n[b,:] . nK[p,:]  via V_WMMA_F32_16X16X4_F32.
// One wave per 16x16 tile; grid exactly covers (2048/16)x(512/16)=4096 tiles,
// so EXEC is all-ones for every WMMA (ISA requirement).
__global__ void cos_sim_wmma(const float* __restrict__ qn, const float* __restrict__ nK,
                             float* __restrict__ cosm) {
  const int tilesN = POOL / 16;               // 32
  int wave = threadIdx.x >> 5;
  int lane = threadIdx.x & 31;
  int tile = blockIdx.x * (blockDim.x >> 5) + wave;
  int tm = tile / tilesN;                     // 0..127
  int tn = tile - tm * tilesN;                // 0..31
  int r  = lane & 15;
  int kh = (lane >> 4) << 1;                  // K sub-offset 0 or 2 per lane half
  const float* aptr = qn + (size_t)(tm * 16 + r) * KEYD + kh;
  const float* bptr = nK + (size_t)(tn * 16 + r) * KEYD + kh;
  v8f acc = {};
#pragma unroll 8
  for (int k = 0; k < KEYD; k += 4) {
    v2f a = *(const v2f*)(aptr + k);
    v2f b = *(const v2f*)(bptr + k);
    acc = __builtin_amdgcn_wmma_f32_16x16x4_f32(
        /*neg_a=*/false, a, /*neg_b=*/false, b,
        /*c_mod=*/(short)0, acc, /*reuse_a=*/false, /*reuse_b=*/false);
  }
  // D layout: VGPR v -> M = v + 8*(lane/16), N = lane%16
  int n  = tn * 16 + r;
  int m0 = tm * 16 + ((lane >> 4) << 3);
#pragma unroll
  for (int v = 0; v < 8; ++v) cosm[(size_t)(m0 + v) * POOL + n] = acc[v];
}

__global__ void zero_counts(float* __restrict__ counts) {
  int p = blockIdx.x * blockDim.x + threadIdx.x;
  if (p < POOL) counts[p] = 0.f;
}

// Per-row top-4 (largest cos == smallest distance) + float bincount.
__global__ void topk_counts(const float* __restrict__ cosm, int* __restrict__ kidx,
                            float* __restrict__ counts) {
  int b = blockIdx.x * blockDim.x + threadIdx.x;
  if (b >= B_SZ) return;
  const float* row = cosm + (size_t)b * POOL;
  float v0 = -1e30f, v1 = -1e30f, v2 = -1e30f, v3 = -1e30f;
  int   i0 = 0, i1 = 0, i2 = 0, i3 = 0;
  for (int p = 0; p < POOL; ++p) {
    float c = row[p];                         // strict '>' => first-index tie-break
    if (c > v0)      { v3=v2;i3=i2; v2=v1;i2=i1; v1=v0;i1=i0; v0=c;i0=p; }
    else if (c > v1) { v3=v2;i3=i2; v2=v1;i2=i1; v1=c;i1=p; }
    else if (c > v2) { v3=v2;i3=i2; v2=c;i2=p; }
    else if (c > v3) { v3=c;i3=p; }
  }
  kidx[b*4+0]=i0; kidx[b*4+1]=i1; kidx[b*4+2]=i2; kidx[b*4+3]=i3;
  atomicAdd(counts + i0, 1.f); atomicAdd(counts + i1, 1.f);
  atomicAdd(counts + i2, 1.f); atomicAdd(counts + i3, 1.f);
}

// colsum[p] = sum_b cos[b][p]   (coalesced across p)
__global__ void col_sums(const float* __restrict__ cosm, float* __restrict__ colsum) {
  int p = blockIdx.x * blockDim.x + threadIdx.x;
  if (p >= POOL) return;
  float s = 0.f;
  for (int b = 0; b < B_SZ; ++b) s += cosm[(size_t)b * POOL + p];
  colsum[p] = s;
}

// loss = (1/(B*K)) * sum_p counts[p] * (1 - colsum[p]/B)
__global__ void loss_reduce(const float* __restrict__ colsum,
                            const float* __restrict__ counts,
                            float* __restrict__ loss) {
  __shared__ float sm[POOL];
  int p = threadIdx.x;
  sm[p] = counts[p] * (1.f - colsum[p] * (1.f / (float)B_SZ));
  __syncthreads();
  for (int s = POOL / 2; s; s >>= 1) {
    if (p < s) sm[p] += sm[p + s];
    __syncthreads();
  }
  if (p == 0) *loss = sm[0] * (1.f / ((float)B_SZ * (float)TOPK));
}

// Pk/Pv assembly: one 192-thread block per (b, r) row; float4 (b128) moves.
// r<16: Ek/Ev gathered from e_p[k_idx]; r>=16: broadcast g_p halves.
__global__ void assemble_prompts(const float* __restrict__ e_p, const float* __restrict__ g_p,
                                 const int* __restrict__ kidx,
                                 float* __restrict__ Pk, float* __restrict__ Pv) {
  int br = blockIdx.x;
  int b  = br / 20;
  int r  = br - b * 20;
  const float4 *srcK, *srcV;
  if (r < 16) {
    int kk = r >> 2, rr = r & 3;
    int idx = kidx[b * 4 + kk];
    srcK = (const float4*)(e_p + ((size_t)idx * EPL + rr) * EMB);
    srcV = (const float4*)(e_p + ((size_t)idx * EPL + 4 + rr) * EMB);
  } else {
    srcK = (const float4*)(g_p + (size_t)(r - 16) * EMB);
    srcV = (const float4*)(g_p + (size_t)(r - 12) * EMB);
  }
  float4* dK = (float4*)(Pk + ((size_t)b * 20 + r) * EMB);
  float4* dV = (float4*)(Pv + ((size_t)b * 20 + r) * EMB);
  int c = threadIdx.x;                        // 0..191 (768/4 float4s)
  dK[c] = srcK[c];
  dV[c] = srcV[c];
}

// x_block passthrough; destination offset is only 4B-aligned -> scalar b32.
__global__ void copy_xblock(const float* __restrict__ src, float* __restrict__ dst, int n) {
  int i = blockIdx.x * blockDim.x + threadIdx.x;
  if (i < n) dst[i] = src[i];
}

extern "C" void kernel_launch(void* const* d_in, const int* in_sizes, int n_in,
                              void* d_out, int out_size, void* d_ws, size_t ws_size,
                              hipStream_t stream) {
  const float* x_querry = (const float*)d_in[0];   // [2048,768]
  const float* x_block  = (const float*)d_in[1];   // [2048,768]
  const float* e_k      = (const float*)d_in[2];   // [512,768]
  const float* e_p      = (const float*)d_in[3];   // [512,8,768]
  const float* g_p      = (const float*)d_in[4];   // [8,768]
  (void)in_sizes; (void)n_in; (void)out_size; (void)ws_size;

  // Workspace partition (floats): qn | nK | cos | colsum | kidx(int)
  float* qn     = (float*)d_ws;
  float* nK     = qn + (size_t)B_SZ * KEYD;
  float* cosm   = nK + (size_t)POOL * KEYD;
  float* colsum = cosm + (size_t)B_SZ * POOL;
  int*   kidx   = (int*)(colsum + POOL);

  // Output partition (floats, reference return order)
  float* Pk     = (float*)d_out;
  float* Pv     = Pk + (size_t)B_SZ * 20 * EMB;
  float* loss   = Pv + (size_t)B_SZ * 20 * EMB;
  float* counts = loss + 1;
  float* xb_out = counts + POOL;

  l2norm_rows<<<B_SZ / 8, 256, 0, stream>>>(x_querry, qn);
  l2norm_rows<<<POOL / 8, 256, 0, stream>>>(e_k, nK);

  // 4096 tiles, 4 waves (tiles) per 128-thread block
  cos_sim_wmma<<<(B_SZ / 16) * (POOL / 16) / 4, 128, 0, stream>>>(qn, nK, cosm);

  zero_counts<<<(POOL + 255) / 256, 256, 0, stream>>>(counts);
  topk_counts<<<B_SZ / 256, 256, 0, stream>>>(cosm, kidx, counts);
  col_sums<<<(POOL + 255) / 256, 256, 0, stream>>>(cosm, colsum);
  loss_reduce<<<1, POOL, 0, stream>>>(colsum, counts, loss);

  assemble_prompts<<<B_SZ * 20, EMB / 4, 0, stream>>>(e_p, g_p, kidx, Pk, Pv);
  copy_xblock<<<(B_SZ * EMB) / 256, 256, 0, stream>>>(x_block, xb_out, B_SZ * EMB);
}